// AmpNorm_60301340836325
// MI455X (gfx1250) — compile-verified
//
#include <hip/hip_runtime.h>
#include <math.h>

// ---------------------------------------------------------------------------
// AmpNorm via DFT-as-matmul on CDNA5 WMMA fp32 (V_WMMA_F32_16X16X4_F32)
//   F = W X W  (W[j,k] = e^{-2pi i jk/512}, symmetric)
//   amp = fftshift(|F|), EMA over batch into running buffer
//   G = (new_amp_ifftshift / |F|) * F        (== new_amp * e^{i*angle(F)})
//   out = Re(W* G W*) / N^2
// Each wave computes a 16x64 output strip (4 N-tiles share the A fragment),
// images are batched per launch via gridDim.y (chunk size from ws_size).
// ---------------------------------------------------------------------------

typedef __attribute__((ext_vector_type(2))) float v2f;
typedef __attribute__((ext_vector_type(8))) float v8f;

#define NDIM 512
#define NNE  (NDIM * NDIM)          // 262144 elements per plane
#define GEMM_BLOCK 256              // 8 wave32
#define GEMM_BLOCKS_X 32            // 256 waves = 32 Mtiles x 8 Nstrips

__device__ __forceinline__ v8f wmma4(v2f a, v2f b, v8f c) {
  // D = A(16x4 f32) * B(4x16 f32) + C(16x16 f32)
  return __builtin_amdgcn_wmma_f32_16x16x4_f32(false, a, false, b,
                                               (short)0, c, false, false);
}

// A-matrix 16x4 fragment (lanes 0-15: K=0/1 in v[0]/v[1]; lanes 16-31: K=2/3)
__device__ __forceinline__ v2f load_a(const float* __restrict__ A, int m0,
                                      int k, int lane) {
  int row = lane & 15;
  int kk  = (lane >> 4) << 1;                  // 0 or 2
  const float* p = A + (size_t)(m0 + row) * NDIM + (k + kk);
  v2f a; a.x = p[0]; a.y = p[1];               // merges to global_load_b64
  return a;
}

// B-matrix 4x16 fragment (v[0]: rows K=0(l0-15)/K=1(l16-31); v[1]: K=2/K=3)
__device__ __forceinline__ v2f load_b(const float* __restrict__ B, int k,
                                      int n0, int lane) {
  int col = n0 + (lane & 15);
  int kb  = lane >> 4;                         // 0 or 1
  v2f b;
  b.x = B[(size_t)(k + kb)     * NDIM + col];
  b.y = B[(size_t)(k + 2 + kb) * NDIM + col];
  return b;
}

// ---- DFT matrix generation: Wr = cos, Wi = -sin (forward twiddles) --------
__global__ void k_gendft(float* __restrict__ Wr, float* __restrict__ Wi) {
  int idx = blockIdx.x * blockDim.x + threadIdx.x;
  if (idx >= NNE) return;
  int j = idx >> 9, k = idx & 511;
  int t = (j * k) & 511;                       // exponent mod N (exact)
  float ang = -2.0f * 3.14159265358979323846f * (float)t * (1.0f / 512.0f);
  float s, c;
  sincosf(ang, &s, &c);
  Wr[idx] = c;
  Wi[idx] = s;
}

// ---- Row DFT (real input): Y = X * (Wr + i*Wi), one image per blockIdx.y --
__global__ void __launch_bounds__(GEMM_BLOCK)
k_rowdft(const float* __restrict__ Xb, const float* __restrict__ Wr,
         const float* __restrict__ Wi, float* __restrict__ Yrb,
         float* __restrict__ Yib) {
  const size_t po = (size_t)blockIdx.y * NNE;
  const float* X  = Xb + po;
  float* Yr = Yrb + po;
  float* Yi = Yib + po;
  int wave = (blockIdx.x * blockDim.x + threadIdx.x) >> 5;  // 0..255
  int lane = threadIdx.x & 31;
  int m0 = (wave >> 3) << 4;                   // 32 M-tiles
  int n0 = (wave & 7) << 6;                    // 8 N-strips of 64
  v8f cr[4] = {}; v8f ci[4] = {};
  for (int k = 0; k < NDIM; k += 4) {
    v2f a = load_a(X, m0, k, lane);            // shared by 4 N-tiles
#pragma unroll
    for (int t = 0; t < 4; ++t) {
      v2f br = load_b(Wr, k, n0 + 16 * t, lane);
      v2f bi = load_b(Wi, k, n0 + 16 * t, lane);
      cr[t] = wmma4(a, br, cr[t]);
      ci[t] = wmma4(a, bi, ci[t]);
    }
  }
  int rb = m0 + ((lane >> 4) << 3);
#pragma unroll
  for (int t = 0; t < 4; ++t) {
    int col = n0 + 16 * t + (lane & 15);
#pragma unroll
    for (int v = 0; v < 8; ++v) {
      size_t idx = (size_t)(rb + v) * NDIM + col;
      Yr[idx] = cr[t][v];
      Yi[idx] = ci[t][v];
    }
  }
}

// ---- Complex GEMM: C = (Wr + i*Wi)(Br + i*Bi); sgn=-1 forward, ------------
// ---- sgn=+1 conjugates W (inverse). B/C per-image via blockIdx.y. ---------
__global__ void __launch_bounds__(GEMM_BLOCK)
k_cgemm(const float* __restrict__ Ar, const float* __restrict__ Ai,
        const float* __restrict__ Brb, const float* __restrict__ Bib,
        float* __restrict__ Crb, float* __restrict__ Cib, float sgn) {
  const size_t po = (size_t)blockIdx.y * NNE;
  const float* Br = Brb + po;
  const float* Bi = Bib + po;
  float* Cr = Crb + po;
  float* Ci = Cib + po;
  int wave = (blockIdx.x * blockDim.x + threadIdx.x) >> 5;
  int lane = threadIdx.x & 31;
  int m0 = (wave >> 3) << 4;
  int n0 = (wave & 7) << 6;
  v8f cr[4] = {}; v8f ci[4] = {};
  for (int k = 0; k < NDIM; k += 4) {
    v2f ar = load_a(Ar, m0, k, lane);
    v2f ai = load_a(Ai, m0, k, lane);
    v2f ais; ais.x =  sgn * ai.x; ais.y =  sgn * ai.y;   // +/-Ai for Cr
    v2f ain; ain.x = -sgn * ai.x; ain.y = -sgn * ai.y;   // -/+Ai for Ci
#pragma unroll
    for (int t = 0; t < 4; ++t) {
      v2f br = load_b(Br, k, n0 + 16 * t, lane);
      v2f bi = load_b(Bi, k, n0 + 16 * t, lane);
      cr[t] = wmma4(ar,  br, cr[t]);
      cr[t] = wmma4(ais, bi, cr[t]);   // Cr = Ar*Br + sgn*Ai*Bi
      ci[t] = wmma4(ar,  bi, ci[t]);
      ci[t] = wmma4(ain, br, ci[t]);   // Ci = Ar*Bi - sgn*Ai*Br
    }
  }
  int rb = m0 + ((lane >> 4) << 3);
#pragma unroll
  for (int t = 0; t < 4; ++t) {
    int col = n0 + 16 * t + (lane & 15);
#pragma unroll
    for (int v = 0; v < 8; ++v) {
      size_t idx = (size_t)(rb + v) * NDIM + col;
      Cr[idx] = cr[t][v];
      Ci[idx] = ci[t][v];
    }
  }
}

// ---- Inverse row transform, real output: out = (Tr*Wr + Ti*Wi) / N^2 ------
__global__ void __launch_bounds__(GEMM_BLOCK)
k_invrow(const float* __restrict__ Trb, const float* __restrict__ Tib,
         const float* __restrict__ Wr, const float* __restrict__ Wi,
         float* __restrict__ outb) {
  const size_t po = (size_t)blockIdx.y * NNE;
  const float* Tr = Trb + po;
  const float* Ti = Tib + po;
  float* out = outb + po;
  int wave = (blockIdx.x * blockDim.x + threadIdx.x) >> 5;
  int lane = threadIdx.x & 31;
  int m0 = (wave >> 3) << 4;
  int n0 = (wave & 7) << 6;
  v8f c[4] = {};
  for (int k = 0; k < NDIM; k += 4) {
    v2f atr = load_a(Tr, m0, k, lane);
    v2f ati = load_a(Ti, m0, k, lane);
#pragma unroll
    for (int t = 0; t < 4; ++t) {
      v2f bwr = load_b(Wr, k, n0 + 16 * t, lane);
      v2f bwi = load_b(Wi, k, n0 + 16 * t, lane);
      c[t] = wmma4(atr, bwr, c[t]);   // Re((Tr+iTi)(Wr-iWi)) = TrWr + TiWi
      c[t] = wmma4(ati, bwi, c[t]);
    }
  }
  const float inv = 1.0f / (float)NNE;
  int rb = m0 + ((lane >> 4) << 3);
#pragma unroll
  for (int t = 0; t < 4; ++t) {
    int col = n0 + 16 * t + (lane & 15);
#pragma unroll
    for (int v = 0; v < 8; ++v)
      out[(size_t)(rb + v) * NDIM + col] = c[t][v] * inv;
  }
}

// ---- Deterministic fftshifted amplitude accumulation over one chunk -------
// One thread owns one (c, us, vs) cell; serial sum over chunk images.
__global__ void k_ampacc(const float* __restrict__ Frb,
                         const float* __restrict__ Fib,
                         float* __restrict__ ampsum, int b0, int g) {
  int idx = blockIdx.x * blockDim.x + threadIdx.x;
  if (idx >= 3 * NNE) return;
  int c   = idx / NNE;
  int pos = idx - c * NNE;
  int us = pos >> 9, vs = pos & 511;
  int u = (us + 256) & 511, v = (vs + 256) & 511;   // shift is an involution
  size_t src = (size_t)u * NDIM + v;
  float s = 0.0f;
  for (int y = 0; y < g; ++y) {
    if (((b0 + y) % 3) == c) {
      float fr = Frb[(size_t)y * NNE + src];
      float fi = Fib[(size_t)y * NNE + src];
      s += sqrtf(fr * fr + fi * fi);
    }
  }
  ampsum[idx] += s;
}

// ---- Amplitude substitution: F *= new_amp_ifftshift / |F| -----------------
__global__ void k_scale(float* __restrict__ Frb, float* __restrict__ Fib,
                        const float* __restrict__ newamp, int b0) {
  int idx = blockIdx.x * blockDim.x + threadIdx.x;
  if (idx >= NNE) return;
  float* Fr = Frb + (size_t)blockIdx.y * NNE;
  float* Fi = Fib + (size_t)blockIdx.y * NNE;
  int c = (b0 + (int)blockIdx.y) % 3;
  const float* na_c = newamp + (size_t)c * NNE;
  int u = idx >> 9, v = idx & 511;
  int us = (u + 256) & 511, vs = (v + 256) & 511;   // ifftshift (even N)
  float na = na_c[(size_t)us * NDIM + vs];
  float fr = Fr[idx], fi = Fi[idx];
  float amp = sqrtf(fr * fr + fi * fi);
  if (amp > 0.0f) {
    float s = na / amp;
    Fr[idx] = fr * s;
    Fi[idx] = fi * s;
  } else {                                          // angle(0)=0 -> e^{i0}=1
    Fr[idx] = na;
    Fi[idx] = 0.0f;
  }
}

// ---- Sum of running_amp (for the zero-initialization branch) --------------
__global__ void k_sum(const float* __restrict__ r, int n,
                      float* __restrict__ out) {
  __shared__ float sdata[256];
  int tid = threadIdx.x;
  float s = 0.0f;
  for (int i = blockIdx.x * blockDim.x + tid; i < n;
       i += gridDim.x * blockDim.x)
    s += r[i];
  sdata[tid] = s;
  __syncthreads();
  for (int off = 128; off > 0; off >>= 1) {
    if (tid < off) sdata[tid] += sdata[tid + off];
    __syncthreads();
  }
  if (tid == 0) atomicAdd(out, sdata[0]);
}

// ---- EMA of running amplitude ---------------------------------------------
__global__ void k_ema(const float* __restrict__ runamp,
                      const float* __restrict__ ampsum,
                      const float* __restrict__ sumflag,
                      float* __restrict__ newamp, int n) {
  int i = blockIdx.x * blockDim.x + threadIdx.x;
  if (i >= n) return;
  float mean = ampsum[i] * (1.0f / 32.0f);
  newamp[i] = (*sumflag == 0.0f) ? mean
                                 : 0.9f * runamp[i] + 0.1f * mean;
}

extern "C" void kernel_launch(void* const* d_in, const int* in_sizes, int n_in,
                              void* d_out, int out_size, void* d_ws,
                              size_t ws_size, hipStream_t stream) {
  const float* x      = (const float*)d_in[0];   // [32,3,512,512]
  const float* runamp = (const float*)d_in[1];   // [3,512,512]
  float* out = (float*)d_out;
  float* ws  = (float*)d_ws;

  const int NIMG = 96;                           // 32 batch * 3 channels

  // Chunk size from ws_size (fixed per session -> deterministic, graph-safe).
  const size_t fixedF = 2 * (size_t)NNE   // Wr, Wi
                      + 6 * (size_t)NNE   // ampsum, newamp (3 planes each)
                      + 16;               // sumflag + pad
  const size_t perImgF = 4 * (size_t)NNE; // Yr, Yi, Fr, Fi
  size_t availF = ws_size / sizeof(float);
  int G = 1;
  if (availF > fixedF + perImgF) {
    size_t g = (availF - fixedF) / perImgF;
    G = (int)(g > 16 ? 16 : g);
    if (G < 1) G = 1;
  }

  // Workspace layout
  float* Wr      = ws;
  float* Wi      = Wr + NNE;
  float* ampsum  = Wi + NNE;                     // 3*NNE
  float* newamp  = ampsum + 3 * (size_t)NNE;     // 3*NNE
  float* sumflag = newamp + 3 * (size_t)NNE;     // 1 (+pad to 16)
  float* Yr      = sumflag + 16;                 // G*NNE (also reused for T)
  float* Yi      = Yr + (size_t)G * NNE;
  float* Fr      = Yi + (size_t)G * NNE;
  float* Fi      = Fr + (size_t)G * NNE;

  hipMemsetAsync(ampsum, 0, 3 * (size_t)NNE * sizeof(float), stream);
  hipMemsetAsync(sumflag, 0, sizeof(float), stream);

  k_gendft<<<NNE / 256, 256, 0, stream>>>(Wr, Wi);

  // Pass 1: forward FFT per chunk, deterministically accumulate amplitudes.
  for (int b0 = 0; b0 < NIMG; b0 += G) {
    int g = (NIMG - b0 < G) ? (NIMG - b0) : G;
    dim3 grid(GEMM_BLOCKS_X, g);
    k_rowdft<<<grid, GEMM_BLOCK, 0, stream>>>(x + (size_t)b0 * NNE, Wr, Wi,
                                              Yr, Yi);
    k_cgemm<<<grid, GEMM_BLOCK, 0, stream>>>(Wr, Wi, Yr, Yi, Fr, Fi, -1.0f);
    k_ampacc<<<(3 * NNE + 255) / 256, 256, 0, stream>>>(Fr, Fi, ampsum, b0, g);
  }

  k_sum<<<96, 256, 0, stream>>>(runamp, 3 * NNE, sumflag);
  k_ema<<<(3 * NNE + 255) / 256, 256, 0, stream>>>(runamp, ampsum, sumflag,
                                                   newamp, 3 * NNE);

  // Pass 2: recompute F, substitute amplitude, inverse FFT, real part.
  for (int b0 = 0; b0 < NIMG; b0 += G) {
    int g = (NIMG - b0 < G) ? (NIMG - b0) : G;
    dim3 grid(GEMM_BLOCKS_X, g);
    dim3 gridS(NNE / 256, g);
    k_rowdft<<<grid, GEMM_BLOCK, 0, stream>>>(x + (size_t)b0 * NNE, Wr, Wi,
                                              Yr, Yi);
    k_cgemm<<<grid, GEMM_BLOCK, 0, stream>>>(Wr, Wi, Yr, Yi, Fr, Fi, -1.0f);
    k_scale<<<gridS, 256, 0, stream>>>(Fr, Fi, newamp, b0);
    // T = conj(W) * G  (into Yr/Yi, dead here)
    k_cgemm<<<grid, GEMM_BLOCK, 0, stream>>>(Wr, Wi, Fr, Fi, Yr, Yi, +1.0f);
    k_invrow<<<grid, GEMM_BLOCK, 0, stream>>>(Yr, Yi, Wr, Wi,
                                              out + (size_t)b0 * NNE);
  }
}